// ParagraphVectorDBOW_32091995636394
// MI455X (gfx1250) — compile-verified
//
#include <hip/hip_runtime.h>
#include <hip/hip_bf16.h>

// ParagraphVector DBOW loss on gfx1250.
// Strategy: L2-resident gather (51.2MB table << 192MB L2). Full-precision
// f32 WMMA 16x16x4: A = 16 gathered token embeddings (float4 gathers,
// each row read exactly once), B = emb_e[b] broadcast across N, held in
// 64 VGPRs per lane and reused for all 96 tiles of the row. Deterministic
// two-phase reduction (per-wave partials in d_ws, fixed-order final sum).

typedef __attribute__((ext_vector_type(2))) float v2f;
typedef __attribute__((ext_vector_type(8))) float v8f;

#define BSZ      1024
#define D        128
#define PAD_LEN  256
#define N_NEG    5
#define NEG_LEN  (PAD_LEN * N_NEG)          // 1280
#define TILES_POS (PAD_LEN / 16)            // 16
#define TILES_NEG (NEG_LEN / 16)            // 80
#define TILES_TOT (TILES_POS + TILES_NEG)   // 96
#define WAVES_PER_BLK 8
#define N_WAVES  (BSZ * WAVES_PER_BLK)      // 8192

__device__ __forceinline__ float log_sigmoid(float x) {
    // stable: min(x,0) - log1p(exp(-|x|))
    return fminf(x, 0.0f) - log1pf(__expf(-fabsf(x)));
}

__global__ void __launch_bounds__(256)
pv_dbow_main(const float* __restrict__ emb_e,
             const float* __restrict__ table,
             const int*   __restrict__ tokens,
             const int*   __restrict__ token_neg,
             float*       __restrict__ lpart,
             int*         __restrict__ cpart) {
    const int b    = blockIdx.x;
    const int wave = threadIdx.x >> 5;
    const int lane = threadIdx.x & 31;
    const int m    = lane & 15;        // token (A-matrix row) this lane feeds
    const int h    = (lane >> 4) & 1;  // lane half: which 64 K-elements

    // Pin this lane's half of emb_e[b] in 64 VGPRs; reused for every tile.
    float bar[64];
    {
        const float* ep = emb_e + b * D + h * 64;
#pragma unroll
        for (int j = 0; j < 16; ++j) {
            float4 e = *(const float4*)(ep + 4 * j);
            bar[4 * j + 0] = e.x; bar[4 * j + 1] = e.y;
            bar[4 * j + 2] = e.z; bar[4 * j + 3] = e.w;
        }
    }

    float lsum = 0.0f;
    int   cnt  = 0;

    for (int t = wave; t < TILES_TOT; t += WAVES_PER_BLK) {
        const int*  idx;
        float       sign;
        if (t < TILES_POS) { idx = tokens    + b * PAD_LEN +  t              * 16; sign =  1.0f; }
        else               { idx = token_neg + b * NEG_LEN + (t - TILES_POS) * 16; sign = -1.0f; }

        const int tok = idx[m];
        if (t < TILES_POS) {
            // count nonzero (non-pad) positive tokens; lanes<16 only (each
            // token appears in two lanes). Ballot is wave-uniform.
            unsigned long long bal = __ballot((tok != 0) && (lane < 16));
            cnt += (int)__popcll(bal);
        }

        const float* rowp = table + (long)tok * D + h * 64;

        v8f c = {};
#pragma unroll
        for (int j = 0; j < 16; ++j) {
            float4 a = *(const float4*)(rowp + 4 * j);  // global_load_b128 gather
            v2f A0 = { a.x, a.y };
            v2f A1 = { a.z, a.w };
            v2f B0 = { bar[4 * j + 0], bar[4 * j + 1] };
            v2f B1 = { bar[4 * j + 2], bar[4 * j + 3] };
            c = __builtin_amdgcn_wmma_f32_16x16x4_f32(false, A0, false, B0,
                                                      (short)0, c, false, false);
            c = __builtin_amdgcn_wmma_f32_16x16x4_f32(false, A1, false, B1,
                                                      (short)0, c, false, false);
        }

        // C VGPR r on lane-half h holds dot for token M = r + 8h (replicated
        // over the 16 lanes of the half).
#pragma unroll
        for (int r = 0; r < 8; ++r) {
            lsum += log_sigmoid(sign * c[r]);
        }
    }

    // Halves hold disjoint token sets; combine lane0-group + lane16-group.
    float other = __shfl_xor(lsum, 16, 32);
    float tot   = lsum + other;

    if (lane == 0) {
        int gw = b * WAVES_PER_BLK + wave;
        lpart[gw] = tot;
        cpart[gw] = cnt;
    }
}

__global__ void __launch_bounds__(256)
pv_dbow_finish(const float* __restrict__ lpart,
               const int*   __restrict__ cpart,
               float*       __restrict__ out) {
    __shared__ float sf[256];
    __shared__ int   si[256];
    const int tid = threadIdx.x;
    float s = 0.0f;
    int   c = 0;
    for (int i = tid; i < N_WAVES; i += 256) { s += lpart[i]; c += cpart[i]; }
    sf[tid] = s; si[tid] = c;
    __syncthreads();
    for (int off = 128; off > 0; off >>= 1) {
        if (tid < off) { sf[tid] += sf[tid + off]; si[tid] += si[tid + off]; }
        __syncthreads();
    }
    if (tid == 0) {
        float n_token = 6.0f * (float)si[0];   // (N_NEG+1) * lens_sum
        out[0] = -sf[0] / n_token;
    }
}

extern "C" void kernel_launch(void* const* d_in, const int* in_sizes, int n_in,
                              void* d_out, int out_size, void* d_ws, size_t ws_size,
                              hipStream_t stream) {
    const float* emb_e     = (const float*)d_in[0];
    const float* emb_table = (const float*)d_in[1];
    const int*   tokens    = (const int*)d_in[2];
    const int*   token_neg = (const int*)d_in[3];

    float* lpart = (float*)d_ws;
    int*   cpart = (int*)((char*)d_ws + N_WAVES * sizeof(float));

    pv_dbow_main<<<BSZ, 256, 0, stream>>>(emb_e, emb_table, tokens, token_neg,
                                          lpart, cpart);
    pv_dbow_finish<<<1, 256, 0, stream>>>(lpart, cpart, (float*)d_out);
}